// Wasserstein1d_33268816675512
// MI455X (gfx1250) — compile-verified
//
#include <hip/hip_runtime.h>

typedef __attribute__((ext_vector_type(2))) float v2f;
typedef __attribute__((ext_vector_type(8))) float v8f;

#define NB  8
#define NT  3000
#define NRC 768                 /* nr*nc = 256*3 */
#define NP  (NB * NRC)          /* 6144 pixels */
#define DT_F 0.001f

// ---------------------------------------------------------------------------
// Pass 1: per-pixel sum of squares over time, on the WMMA path.
// One wave32 owns 16 consecutive pixels (tiles never cross a batch: 768%16==0).
// A-matrix (16x4 f32): lane l holds A[m=l&15, k = 2*(l>>4) + {0,1}] in 2 VGPRs.
// B = all-ones 4x16  =>  C[m,n] accumulates the row sum of x^2 (replicated in n).
// C layout: VGPR r holds row m=r (lanes 0-15) and m=r+8 (lanes 16-31).
// ---------------------------------------------------------------------------
__global__ __launch_bounds__(128) void wsum_kernel(const float* __restrict__ x,
                                                   const float* __restrict__ y,
                                                   float* __restrict__ sx,
                                                   float* __restrict__ sy) {
  const int wave = (blockIdx.x * blockDim.x + threadIdx.x) >> 5;
  const int lane = threadIdx.x & 31;
  const int p0   = wave * 16;
  const int m    = lane & 15;
  const int koff = (lane >> 4) << 1;            // 0 or 2
  const int p    = p0 + m;
  const int b    = p / NRC;
  const int q    = p - b * NRC;
  const float* bx = x + (size_t)(b * NT) * NRC + q;
  const float* by = y + (size_t)(b * NT) * NRC + q;

  v8f cx = {};
  v8f cy = {};
  const v2f ones = {1.f, 1.f};

  for (int k = 0; k < NT; k += 4) {
    const int k0 = k + koff;
    const float xa = bx[(size_t)k0 * NRC];
    const float xb = bx[(size_t)(k0 + 1) * NRC];
    const float ya = by[(size_t)k0 * NRC];
    const float yb = by[(size_t)(k0 + 1) * NRC];
    v2f ax = {xa * xa, xb * xb};
    v2f ay = {ya * ya, yb * yb};
    cx = __builtin_amdgcn_wmma_f32_16x16x4_f32(false, ax, false, ones,
                                               (short)0, cx, false, false);
    cy = __builtin_amdgcn_wmma_f32_16x16x4_f32(false, ay, false, ones,
                                               (short)0, cy, false, false);
  }

  if ((lane & 15) == 0) {                       // lanes 0 and 16 write 8 rows each
    const int pb = p0 + ((lane >> 4) << 3);
#pragma unroll
    for (int r = 0; r < 8; ++r) {
      sx[pb + r] = cx[r];
      sy[pb + r] = cy[r];
    }
  }
}

// ---------------------------------------------------------------------------
// Pass 2: streaming two-pointer quantile merge, one thread per pixel.
// Equivalent to: qs = sort(concat(cx,cy)); ix = searchsorted(cx,qs,'left').clip;
// cost = sum(diff(qs) * (t[ix]-t[iy])^2).  Ties give zero-delta segments.
// Depth-8 per-side register FIFOs keep ~16 loads in flight per thread to hide
// L2 latency (the whole 147.5 MB dataset is L2-resident after pass 1).
// Prefetch addresses clamp to NT-1; clamped garbage is never consumed because
// the cumulative value switches to +inf once a side is exhausted.
// ---------------------------------------------------------------------------
__global__ __launch_bounds__(256) void wmerge_kernel(const float* __restrict__ x,
                                                     const float* __restrict__ y,
                                                     const float* __restrict__ sx,
                                                     const float* __restrict__ sy,
                                                     float* __restrict__ pc) {
  const int p = blockIdx.x * blockDim.x + threadIdx.x;
  if (p >= NP) return;
  const float sxv = sx[p];
  const float syv = sy[p];
  if (sxv == 0.f || syv == 0.f) { pc[p] = 0.f; return; }   // reference skips these
  const float isx = 1.f / sxv;
  const float isy = 1.f / syv;

  const int b = p / NRC;
  const int q = p - b * NRC;
  const float* bx = x + (size_t)(b * NT) * NRC + q;
  const float* by = y + (size_t)(b * NT) * NRC + q;

  auto ldx = [&](int k) -> float {
    const int kc = (k < NT - 1) ? k : (NT - 1);
    const float v = bx[(size_t)kc * NRC];
    return v * v * isx;
  };
  auto ldy = [&](int k) -> float {
    const int kc = (k < NT - 1) ? k : (NT - 1);
    const float v = by[(size_t)kc * NRC];
    return v * v * isy;
  };

  float a  = ldx(0);                 // cx[i] : cumulative CDF candidate of x
  float bc = ldy(0);                 // cy[j]
  float fx0 = ldx(1), fx1 = ldx(2), fx2 = ldx(3), fx3 = ldx(4),
        fx4 = ldx(5), fx5 = ldx(6), fx6 = ldx(7), fx7 = ldx(8);
  float fy0 = ldy(1), fy1 = ldy(2), fy2 = ldy(3), fy3 = ldy(4),
        fy4 = ldy(5), fy5 = ldy(6), fy6 = ldy(7), fy7 = ldy(8);

  int i = 0, j = 0;
  float qprev = 0.f;
  float cost  = 0.f;

  for (int step = 0; step < 2 * NT; ++step) {
    const bool  takeX = (a <= bc);
    const float qv = takeX ? a : bc;
    const int   ii = (i < NT - 1) ? i : (NT - 1);
    const int   jj = (j < NT - 1) ? j : (NT - 1);
    const float td = (float)(ii - jj) * DT_F;
    cost += (qv - qprev) * td * td;
    qprev = qv;
    if (takeX) {
      ++i;
      a = (i < NT) ? (a + fx0) : __builtin_inff();
      fx0 = fx1; fx1 = fx2; fx2 = fx3; fx3 = fx4;
      fx4 = fx5; fx5 = fx6; fx6 = fx7;
      fx7 = ldx(i + 8);
    } else {
      ++j;
      bc = (j < NT) ? (bc + fy0) : __builtin_inff();
      fy0 = fy1; fy1 = fy2; fy2 = fy3; fy3 = fy4;
      fy4 = fy5; fy5 = fy6; fy6 = fy7;
      fy7 = ldy(j + 8);
    }
  }
  pc[p] = cost;
}

// ---------------------------------------------------------------------------
// Pass 3: deterministic fixed-order reduction of 6144 per-pixel costs.
// ---------------------------------------------------------------------------
__global__ __launch_bounds__(256) void wreduce_kernel(const float* __restrict__ pc,
                                                      float* __restrict__ out) {
  __shared__ float s[256];
  const int tid = threadIdx.x;
  float acc = 0.f;
  for (int idx = tid; idx < NP; idx += 256) acc += pc[idx];
  s[tid] = acc;
  __syncthreads();
  for (int off = 128; off > 0; off >>= 1) {
    if (tid < off) s[tid] += s[tid + off];
    __syncthreads();
  }
  if (tid == 0) out[0] = s[0];
}

extern "C" void kernel_launch(void* const* d_in, const int* in_sizes, int n_in,
                              void* d_out, int out_size, void* d_ws, size_t ws_size,
                              hipStream_t stream) {
  (void)in_sizes; (void)n_in; (void)out_size; (void)ws_size;
  const float* x = (const float*)d_in[0];
  const float* y = (const float*)d_in[1];
  float* out = (float*)d_out;
  float* ws  = (float*)d_ws;
  float* sx  = ws;                 // NP floats
  float* sy  = ws + NP;            // NP floats
  float* pc  = ws + 2 * NP;        // NP floats  (73.7 KB total workspace)

  // 384 tiles of 16 pixels, 4 waves per block -> 96 blocks of 128 threads.
  wsum_kernel<<<dim3(96), dim3(128), 0, stream>>>(x, y, sx, sy);
  // One thread per pixel: 6144 threads.
  wmerge_kernel<<<dim3(NP / 256), dim3(256), 0, stream>>>(x, y, sx, sy, pc);
  // Single-block deterministic reduction.
  wreduce_kernel<<<dim3(1), dim3(256), 0, stream>>>(pc, out);
}